// H_13048110646034
// MI455X (gfx1250) — compile-verified
//
#include <hip/hip_runtime.h>

// Butterfly-permuted block-diagonal complex matmul for MI455X (gfx1250).
//
// Roofline: 1.07 GFLOP vs 288 MB traffic (weights 256 MB dominate, each byte
// used once) -> bandwidth bound (~12.4 us at 23.3 TB/s). So: fp32 WMMA
// (v_wmma_f32_16x16x4_f32) to keep math off the critical path, weights
// streamed once with coalesced loads directly into B fragments, x staged via
// LDS and broadcast to all waves, complex combine folded into K ([xr|xi] x
// [W_a;W_b], K=128).

typedef float v2f __attribute__((ext_vector_type(2)));
typedef float v8f __attribute__((ext_vector_type(8)));

namespace {
constexpr int kN      = 16384;  // spatial length (4^7)
constexpr int kR      = 16;     // features
constexpr int kP      = 4096;   // local blocks (N/4)
constexpr int kB      = 8;      // batch
constexpr int kStride = 132;    // LDS row stride in dwords (padded: 8B-aligned,
                                // 132 mod 64 = 4 -> conflict-free b64 frags)
}

__global__ __launch_bounds__(256)
void bfly_wmma_f32(const float* __restrict__ x,
                   const float* __restrict__ Wrr,
                   const float* __restrict__ Wri,
                   const float* __restrict__ Wir,
                   const float* __restrict__ Wii,
                   const int*   __restrict__ perm,
                   float*       __restrict__ out)
{
  __shared__ __align__(16) float xA[kB * kStride];  // rows b=0..7, cols [xr|xi] 0..127

  const int p    = blockIdx.x;      // butterfly block
  const int tid  = threadIdx.x;     // 0..255 (8 wave32)
  const int wave = tid >> 5;
  const int lane = tid & 31;

  // ---- Stage gathered x into LDS: xA[b][i] = x[b, c, perm[4p + (i%64)/16], i%16],
  //      c = 0 for i<64 (xr), c = 1 for i>=64 (xi). 1024 dwords via 256 b128 loads.
  {
    const int seg = tid >> 2;          // 0..63 : (b, c, row-in-window)
    const int sub = tid & 3;           // 4 floats each
    const int b_  = seg >> 3;
    const int c_  = (seg >> 2) & 1;
    const int row = seg & 3;
    const int n_  = perm[4 * p + row];
    const float4 v = *reinterpret_cast<const float4*>(
        x + ((size_t)(b_ * 2 + c_) * kN + (size_t)n_) * kR + sub * 4);
    *reinterpret_cast<float4*>(
        &xA[b_ * kStride + c_ * 64 + row * 16 + sub * 4]) = v;
  }
  __syncthreads();

  // ---- Wave task: (o-tile, component). Waves 0-3: y_re tiles 0-3; 4-7: y_im.
  const int otile = wave & 3;
  const int comp  = wave >> 2;                 // 0 = real, 1 = imag
  const float* WA = comp ? Wir : Wrr;          // K = 0..63   (xr part)
  const float* WB = comp ? Wii : Wri;          // K = 64..127 (xi part)

  // WMMA f32 16x16x4 fragment coordinates (per ISA 7.12.2):
  //   A (16x4): lane = {h = K-half, m = M}; v2f = {A[m][k0+2h], A[m][k0+2h+1]}
  //   B (4x16): lane = {h, n = N};          v2f = {B[k0+2h][n], B[k0+2h+1][n]}
  //   C (16x16): VGPR r holds M=r (lanes 0-15) / M=r+8 (lanes 16-31), N=lane&15
  const int m = lane & 15;                     // also n for the B fragment
  const int h = lane >> 4;

  const float* ar = xA + (m & 7) * kStride + 2 * h;   // rows 8-15 duplicate (unused C rows)
  const float* wa = WA + (size_t)p * 4096 + (size_t)(2 * h) * 64 + otile * 16 + m;
  const float* wb = WB + (size_t)p * 4096 + (size_t)(2 * h) * 64 + otile * 16 + m;

  v8f acc = {0.f, 0.f, 0.f, 0.f, 0.f, 0.f, 0.f, 0.f};

  // K = 0..63 : xr x WA
#pragma unroll
  for (int k0 = 0; k0 < 64; k0 += 4) {
    v2f a = *reinterpret_cast<const v2f*>(ar + k0);   // ds_load_b64
    v2f b;
    b.x = wa[(size_t)k0 * 64];                        // coalesced 64B/half-wave
    b.y = wa[(size_t)k0 * 64 + 64];
    acc = __builtin_amdgcn_wmma_f32_16x16x4_f32(
        false, a, false, b, (short)0, acc, false, false);
  }
  // K = 64..127 : xi x WB
#pragma unroll
  for (int k0 = 0; k0 < 64; k0 += 4) {
    v2f a = *reinterpret_cast<const v2f*>(ar + 64 + k0);
    v2f b;
    b.x = wb[(size_t)k0 * 64];
    b.y = wb[(size_t)k0 * 64 + 64];
    acc = __builtin_amdgcn_wmma_f32_16x16x4_f32(
        false, a, false, b, (short)0, acc, false, false);
  }

  // ---- Store: out flat index = (b*2 + c)*N*R + p*64 + o. C rows 0-7 = batch,
  //      rows 8-15 (lanes 16-31) are padding -> masked (divergence is legal
  //      here: all WMMAs are done).
  if (lane < 16) {
    float* ob = out + (size_t)p * 64 + otile * 16 + lane;
#pragma unroll
    for (int r = 0; r < kB; ++r) {
      ob[(size_t)(r * 2 + comp) * ((size_t)kN * kR)] = acc[r];
    }
  }
}

extern "C" void kernel_launch(void* const* d_in, const int* in_sizes, int n_in,
                              void* d_out, int out_size, void* d_ws, size_t ws_size,
                              hipStream_t stream) {
  const float* x    = (const float*)d_in[0];
  const float* Wrr  = (const float*)d_in[1];
  const float* Wri  = (const float*)d_in[2];
  const float* Wir  = (const float*)d_in[3];
  const float* Wii  = (const float*)d_in[4];
  const int*   perm = (const int*)d_in[5];
  float*       out  = (float*)d_out;

  bfly_wmma_f32<<<dim3(kP), dim3(256), 0, stream>>>(x, Wrr, Wri, Wir, Wii, perm, out);
}